// CrfRnnLayerSP_66168266162186
// MI455X (gfx1250) — compile-verified
//
#include <hip/hip_runtime.h>
#include <math.h>

// Problem constants (match reference)
#define HH   512
#define WWID 512
#define CC   21
#define HWs  (HH * WWID)      // 262144
#define CHWs (CC * HWs)       // 5505024
#define RAD  3                // bilateral radius (49 taps)
#define GR   6                // gaussian radius (theta_gamma=3 -> 2*3=6, 13 taps)

#define TD   16               // bilateral tile dim (16x16 pixels per block)
#define TP   (TD + 2 * RAD)   // 22 (tile + halo)
#define TILE_ELEMS (TP * TP)  // 484

typedef __attribute__((ext_vector_type(2))) float v2f;
typedef __attribute__((ext_vector_type(8))) float v8f;

#if defined(__has_builtin)
#if __has_builtin(__builtin_amdgcn_global_load_async_to_lds_b32)
#define HAVE_ASYNC_LDS 1
typedef __attribute__((address_space(1))) int as1_int;
typedef __attribute__((address_space(3))) int as3_int;
#endif
#endif

// ---------------------------------------------------------------------------
// k_setup_mats: M1 = compat @ Ws, M2 = compat @ Wb, zero-padded to 32x32
// ---------------------------------------------------------------------------
__global__ void k_setup_mats(const float* __restrict__ compat,
                             const float* __restrict__ Ws,
                             const float* __restrict__ Wb,
                             float* __restrict__ M1p, float* __restrict__ M2p) {
    int idx = blockIdx.x * 256 + threadIdx.x;   // 1024 total
    if (idx >= 1024) return;
    int i = idx >> 5, j = idx & 31;
    float s1 = 0.f, s2 = 0.f;
    if (i < CC && j < CC) {
        for (int t = 0; t < CC; ++t) {
            float c = compat[i * CC + t];
            s1 += c * Ws[t * CC + j];
            s2 += c * Wb[t * CC + j];
        }
    }
    M1p[idx] = s1;
    M2p[idx] = s2;
}

// ---------------------------------------------------------------------------
// k_init: transpose unaries (H,W,C) -> channel-major u_t (C,H,W); q = u_t
// ---------------------------------------------------------------------------
__global__ void k_init(const float* __restrict__ un,
                       float* __restrict__ u_t, float* __restrict__ q) {
    int pix = blockIdx.x * 256 + threadIdx.x;
    #pragma unroll
    for (int c = 0; c < CC; ++c) {
        float v = un[pix * CC + c];
        u_t[c * HWs + pix] = v;
        q[c * HWs + pix]   = v;
    }
}

// ---------------------------------------------------------------------------
// k_bilat_weights: per-pixel 49 bilateral weights (tap-major) + norm
// im[c][h][w] = rgb[(h*W+w)*3 + c]; out-of-bounds neighbor treated as 0 for
// the color diff (zero padding), excluded from norm (ones are zero-padded).
// ---------------------------------------------------------------------------
__global__ __launch_bounds__(256) void k_bilat_weights(
        const float* __restrict__ rgb,
        float* __restrict__ bw, float* __restrict__ bnorm) {
    int pix = blockIdx.x * 256 + threadIdx.x;
    int h = pix >> 9, w = pix & 511;
    float r0 = rgb[pix * 3 + 0];
    float g0 = rgb[pix * 3 + 1];
    float b0 = rgb[pix * 3 + 2];
    const float inv2a = 1.f / (2.f * 160.f * 160.f);   // theta_alpha
    const float inv2b = 1.f / (2.f * 3.f * 3.f);       // theta_beta
    float norm = 0.f;
    int k = 0;
    for (int dy = -RAD; dy <= RAD; ++dy) {
        for (int dx = -RAD; dx <= RAD; ++dx, ++k) {
            int hs = h + dy, wx = w + dx;
            bool inb = (hs >= 0 && hs < HH && wx >= 0 && wx < WWID);
            float r = 0.f, g = 0.f, b = 0.f;
            if (inb) {
                int np = hs * WWID + wx;
                r = rgb[np * 3 + 0];
                g = rgb[np * 3 + 1];
                b = rgb[np * 3 + 2];
            }
            float dr = r0 - r, dg = g0 - g, db = b0 - b;
            float d2 = dr * dr + dg * dg + db * db;
            float wt = expf(-(float)(dx * dx + dy * dy) * inv2a - d2 * inv2b);
            bw[k * HWs + pix] = wt;
            norm += inb ? wt : 0.f;
        }
    }
    bnorm[pix] = norm;
}

// ---------------------------------------------------------------------------
// k_softmax: sm = softmax over channel axis of q
// ---------------------------------------------------------------------------
__global__ __launch_bounds__(256) void k_softmax(const float* __restrict__ q,
                                                 float* __restrict__ sm) {
    int pix = blockIdx.x * 256 + threadIdx.x;
    float v[CC];
    float m = -3.402823e38f;
    #pragma unroll
    for (int c = 0; c < CC; ++c) {
        v[c] = q[c * HWs + pix];
        m = fmaxf(m, v[c]);
    }
    float s = 0.f;
    #pragma unroll
    for (int c = 0; c < CC; ++c) {
        v[c] = expf(v[c] - m);
        s += v[c];
    }
    float inv = 1.f / s;
    #pragma unroll
    for (int c = 0; c < CC; ++c) sm[c * HWs + pix] = v[c] * inv;
}

// ---------------------------------------------------------------------------
// k_gauss_v / k_gauss_h: separable 13-tap gaussian (theta=3, zero-pad SAME);
// horizontal pass also divides by the separable norm rowsum(h)*colsum(w).
// ---------------------------------------------------------------------------
__global__ __launch_bounds__(256) void k_gauss_v(const float* __restrict__ in,
                                                 float* __restrict__ out) {
    int idx = blockIdx.x * 256 + threadIdx.x;   // over C*H*W
    int c = idx / HWs;
    int rem = idx - c * HWs;
    int h = rem >> 9, w = rem & 511;
    float acc = 0.f;
    #pragma unroll
    for (int d = -GR; d <= GR; ++d) {
        int hs = h + d;
        if (hs >= 0 && hs < HH) {
            float kc = expf(-(float)(d * d) / 18.f);
            acc += kc * in[c * HWs + hs * WWID + w];
        }
    }
    out[idx] = acc;
}

__global__ __launch_bounds__(256) void k_gauss_h(const float* __restrict__ in,
                                                 float* __restrict__ out) {
    int idx = blockIdx.x * 256 + threadIdx.x;
    int c = idx / HWs;
    int rem = idx - c * HWs;
    int h = rem >> 9, w = rem & 511;
    float acc = 0.f, rs = 0.f, cs = 0.f;
    #pragma unroll
    for (int d = -GR; d <= GR; ++d) {
        float kc = expf(-(float)(d * d) / 18.f);
        int wx = w + d;
        if (wx >= 0 && wx < WWID) {
            acc += kc * in[c * HWs + h * WWID + wx];
            cs += kc;
        }
        int hs = h + d;
        if (hs >= 0 && hs < HH) rs += kc;
    }
    out[idx] = acc / (rs * cs);
}

// ---------------------------------------------------------------------------
// k_bilat_apply: 49-tap bilateral filtering of all 21 channels, / norm.
// LDS-tiled: each 256-thread block owns a 16x16 pixel tile; the 22x22x21
// sm halo (40.6 KB) is staged into LDS via GLOBAL_LOAD_ASYNC_TO_LDS (ASYNCcnt)
// when available, then consumed with conflict-free ds_loads. OOB halo slots
// are zeroed so the 49-tap accumulate needs no bounds branches (matches the
// reference's zero padding; the norm counts in-bounds taps only).
// ---------------------------------------------------------------------------
__global__ __launch_bounds__(256) void k_bilat_apply(
        const float* __restrict__ sm, const float* __restrict__ bw,
        const float* __restrict__ bnorm, float* __restrict__ out) {
    __shared__ float tile[CC * TILE_ELEMS];   // 21*484*4 = 40656 B

    int tid = threadIdx.x;
    int by0 = (blockIdx.x >> 5) * TD;   // 32x32 tiles
    int bx0 = (blockIdx.x & 31) * TD;

    // Cooperative halo fill: each LDS slot written exactly once.
    for (int e = tid; e < CC * TILE_ELEMS; e += 256) {
        int c = e / TILE_ELEMS;
        int rem = e - c * TILE_ELEMS;
        int r = rem / TP, cl = rem - r * TP;
        int gh = by0 + r - RAD, gw = bx0 + cl - RAD;
        if (gh >= 0 && gh < HH && gw >= 0 && gw < WWID) {
#ifdef HAVE_ASYNC_LDS
            __builtin_amdgcn_global_load_async_to_lds_b32(
                (as1_int*)(sm + (size_t)c * HWs + gh * WWID + gw),
                (as3_int*)&tile[e], 0, 0);
#else
            tile[e] = sm[(size_t)c * HWs + gh * WWID + gw];
#endif
        } else {
            tile[e] = 0.f;
        }
    }
#ifdef HAVE_ASYNC_LDS
#if __has_builtin(__builtin_amdgcn_s_wait_asynccnt)
    __builtin_amdgcn_s_wait_asynccnt(0);
#else
    asm volatile("s_wait_asynccnt 0x0" ::: "memory");
#endif
#endif
    __syncthreads();

    int tx = tid & 15, ty = tid >> 4;
    int pix = (by0 + ty) * WWID + (bx0 + tx);

    float acc[CC];
    #pragma unroll
    for (int c = 0; c < CC; ++c) acc[c] = 0.f;

    for (int k = 0; k < 49; ++k) {
        float wt = bw[k * HWs + pix];
        int dy = k / 7, dx = k - dy * 7;              // 0..6
        int base = (ty + dy) * TP + (tx + dx);        // halo-relative
        #pragma unroll
        for (int c = 0; c < CC; ++c) acc[c] += wt * tile[c * TILE_ELEMS + base];
    }

    float inv = 1.f / bnorm[pix];
    #pragma unroll
    for (int c = 0; c < CC; ++c) out[c * HWs + pix] = acc[c] * inv;
}

// ---------------------------------------------------------------------------
// Superpixel logsumexp: B[c] = log( sum_hw (cond ? exp(sm) : 1) )
// (values in (0,1], no overflow; reference's max-shifted logsumexp is equal)
// Note: cond uses TRANSPOSED sp_map per reference: spm[h][w]=sp_map[w*W+h]
// ---------------------------------------------------------------------------
__global__ __launch_bounds__(256) void k_sp_partial(
        const float* __restrict__ sm, const int* __restrict__ sp_map,
        const int* __restrict__ sp_idx, int iter, float* __restrict__ partial) {
    int c = blockIdx.y;
    int chunk = blockIdx.x;      // 256 chunks of 1024 pixels
    int tid = threadIdx.x;
    int s = sp_idx[iter];
    float local = 0.f;
    #pragma unroll
    for (int t = 0; t < 4; ++t) {
        int pix = chunk * 1024 + t * 256 + tid;
        int h = pix >> 9, w = pix & 511;
        bool cond = (sp_map[w * WWID + h] == s);
        float x = sm[c * HWs + pix];
        local += cond ? expf(x) : 1.0f;
    }
    __shared__ float red[256];
    red[tid] = local;
    __syncthreads();
    for (int o = 128; o > 0; o >>= 1) {
        if (tid < o) red[tid] += red[tid + o];
        __syncthreads();
    }
    if (tid == 0) partial[c * 256 + chunk] = red[0];
}

__global__ __launch_bounds__(256) void k_sp_final(
        const float* __restrict__ partial, float* __restrict__ Bc) {
    int c = blockIdx.x;
    int tid = threadIdx.x;
    __shared__ float red[256];
    red[tid] = partial[c * 256 + tid];
    __syncthreads();
    for (int o = 128; o > 0; o >>= 1) {
        if (tid < o) red[tid] += red[tid + o];
        __syncthreads();
    }
    if (tid == 0) Bc[c] = logf(red[0]);
}

// ---------------------------------------------------------------------------
// k_wmma_update: q = u - (M1p @ spat + M2p @ bil) - sp_upd  (fused epilogue)
// Each wave owns 16 pixels; M,K padded 21->32. Uses V_WMMA_F32_16X16X4_F32:
//   A 16x4 f32 (2 VGPR): lane<16 -> (M=lane, K=0/1); lane>=16 -> (M, K=2/3)
//   B 4x16  f32 (2 VGPR): lane<16 -> (K=0/1, N=lane); lane>=16 -> (K=2/3)
//   D 16x16 f32 (8 VGPR): reg r -> M = r + 8*(lane>=16), N = lane&15
// ---------------------------------------------------------------------------
__device__ __forceinline__ void gemm_acc(const float* __restrict__ in,
                                         const float* __restrict__ M,
                                         int p0, int n, int half,
                                         v8f& acc0, v8f& acc1) {
    #pragma unroll
    for (int kk = 0; kk < 8; ++kk) {
        int c0 = kk * 4 + half * 2;
        v2f b;
        b.x = (c0     < CC) ? in[c0 * HWs + p0 + n]       : 0.f;
        b.y = (c0 + 1 < CC) ? in[(c0 + 1) * HWs + p0 + n] : 0.f;
        v2f a0, a1;
        a0.x = M[n * 32 + c0];
        a0.y = M[n * 32 + c0 + 1];
        a1.x = M[(16 + n) * 32 + c0];
        a1.y = M[(16 + n) * 32 + c0 + 1];
        acc0 = __builtin_amdgcn_wmma_f32_16x16x4_f32(
                   false, a0, false, b, (short)0, acc0, false, false);
        acc1 = __builtin_amdgcn_wmma_f32_16x16x4_f32(
                   false, a1, false, b, (short)0, acc1, false, false);
    }
}

__global__ __launch_bounds__(256) void k_wmma_update(
        const float* __restrict__ spat, const float* __restrict__ bil,
        const float* __restrict__ M1p,  const float* __restrict__ M2p,
        const float* __restrict__ u_t,  const float* __restrict__ sm,
        const int*   __restrict__ sp_map, const int* __restrict__ sp_idx,
        int iter, const float* __restrict__ Bc,
        const float* __restrict__ low_w, const float* __restrict__ high_w_p,
        float* __restrict__ q) {
    int wave = (blockIdx.x * 256 + threadIdx.x) >> 5;   // 16384 waves
    int lane = threadIdx.x & 31;
    int n = lane & 15, half = lane >> 4;
    int p0 = wave * 16;

    v8f acc0 = {0.f, 0.f, 0.f, 0.f, 0.f, 0.f, 0.f, 0.f};
    v8f acc1 = {0.f, 0.f, 0.f, 0.f, 0.f, 0.f, 0.f, 0.f};

    gemm_acc(spat, M1p, p0, n, half, acc0, acc1);
    gemm_acc(bil,  M2p, p0, n, half, acc0, acc1);

    // Fused epilogue: q = u - pairwise - superpixel_update
    int s = sp_idx[iter];
    float highw = high_w_p[0];
    int pix = p0 + n;
    int h = pix >> 9, w = pix & 511;
    bool cond = (sp_map[w * WWID + h] == s);   // transposed sp_map (reference)

    #pragma unroll
    for (int r = 0; r < 8; ++r) {
        {   // M-tile 0: rows 0..15
            int m = 8 * half + r;
            if (m < CC) {
                float smv = sm[m * HWs + pix];
                float qm = (smv == 0.f) ? 1.f : smv;
                float first = cond ? (Bc[m] / qm) : 0.f;
                float upd = low_w[m] * first + highw * (1.f - first);
                q[m * HWs + pix] = u_t[m * HWs + pix] - acc0[r] - upd;
            }
        }
        {   // M-tile 1: rows 16..31 (only 16..20 valid)
            int m = 16 + 8 * half + r;
            if (m < CC) {
                float smv = sm[m * HWs + pix];
                float qm = (smv == 0.f) ? 1.f : smv;
                float first = cond ? (Bc[m] / qm) : 0.f;
                float upd = low_w[m] * first + highw * (1.f - first);
                q[m * HWs + pix] = u_t[m * HWs + pix] - acc1[r] - upd;
            }
        }
    }
}

// ---------------------------------------------------------------------------
// k_out: transpose (C,H,W) -> (H,W,C)
// ---------------------------------------------------------------------------
__global__ void k_out(const float* __restrict__ q, float* __restrict__ out) {
    int pix = blockIdx.x * 256 + threadIdx.x;
    #pragma unroll
    for (int c = 0; c < CC; ++c) out[pix * CC + c] = q[c * HWs + pix];
}

// ---------------------------------------------------------------------------
extern "C" void kernel_launch(void* const* d_in, const int* in_sizes, int n_in,
                              void* d_out, int out_size, void* d_ws, size_t ws_size,
                              hipStream_t stream) {
    const float* unaries = (const float*)d_in[0];   // (1,H,W,C)
    const float* rgb     = (const float*)d_in[1];   // (1,H,W,3)
    const int*   sp_map  = (const int*)  d_in[2];   // (1,H,W)
    const int*   sp_idx  = (const int*)  d_in[3];   // (5,1)
    const float* Ws      = (const float*)d_in[4];   // (C,C)
    const float* Wb      = (const float*)d_in[5];   // (C,C)
    const float* compat  = (const float*)d_in[6];   // (C,C)
    const float* low_w   = (const float*)d_in[7];   // (C,)
    const float* high_w  = (const float*)d_in[8];   // (1,)
    float* out = (float*)d_out;

    // Workspace carve-up (floats)
    float* ws  = (float*)d_ws;
    float* u_t   = ws;                 // CHW
    float* q     = u_t + CHWs;         // CHW
    float* sm    = q + CHWs;           // CHW
    float* spat  = sm + CHWs;          // CHW
    float* bil   = spat + CHWs;        // CHW (doubles as vertical-pass tmp)
    float* bw    = bil + CHWs;         // 49*HW
    float* bnorm = bw + 49 * (size_t)HWs;   // HW
    float* M1p   = bnorm + HWs;        // 1024
    float* M2p   = M1p + 1024;         // 1024
    float* part  = M2p + 1024;         // 21*256
    float* Bc    = part + CC * 256;    // 21

    const int PIXB = HWs / 256;        // 1024 blocks over pixels
    const int ELTB = CHWs / 256;       // 21504 blocks over elements

    // Iteration-invariant precompute
    k_setup_mats<<<4, 256, 0, stream>>>(compat, Ws, Wb, M1p, M2p);
    k_init<<<PIXB, 256, 0, stream>>>(unaries, u_t, q);
    k_bilat_weights<<<PIXB, 256, 0, stream>>>(rgb, bw, bnorm);

    for (int it = 0; it < 5; ++it) {
        k_softmax<<<PIXB, 256, 0, stream>>>(q, sm);
        k_gauss_v<<<ELTB, 256, 0, stream>>>(sm, bil);       // bil as tmp
        k_gauss_h<<<ELTB, 256, 0, stream>>>(bil, spat);
        k_bilat_apply<<<PIXB, 256, 0, stream>>>(sm, bw, bnorm, bil);
        k_sp_partial<<<dim3(256, CC), 256, 0, stream>>>(sm, sp_map, sp_idx, it, part);
        k_sp_final<<<CC, 256, 0, stream>>>(part, Bc);
        k_wmma_update<<<2048, 256, 0, stream>>>(spat, bil, M1p, M2p, u_t, sm,
                                                sp_map, sp_idx, it, Bc,
                                                low_w, high_w, q);
    }
    k_out<<<PIXB, 256, 0, stream>>>(q, out);
}